// GATModel_38860864094519
// MI455X (gfx1250) — compile-verified
//
#include <hip/hip_runtime.h>
#include <math.h>

#define NN 30000
#define NE 480000
#define NG 64
#define NEG_SLOPE 0.2f

typedef __attribute__((ext_vector_type(2))) float v2f;
typedef __attribute__((ext_vector_type(8))) float v8f;

// ---------------------------------------------------------------- utilities

__global__ void fill_kernel(float* __restrict__ p, float v, int n) {
    int i = blockIdx.x * blockDim.x + threadIdx.x;
    if (i < n) p[i] = v;
}

// zero-pad [N,6] -> [N,8] so the GEMM needs no K-tail predication
__global__ void pad_k6_to_k8(const float* __restrict__ in, float* __restrict__ out, int N) {
    int idx = blockIdx.x * blockDim.x + threadIdx.x;
    if (idx >= N * 8) return;
    int n = idx >> 3, k = idx & 7;
    out[idx] = (k < 6) ? in[n * 6 + k] : 0.f;
}

__device__ __forceinline__ float lrelu(float v) {
    return v > 0.f ? v : NEG_SLOPE * v;
}

// float atomic max via signed/unsigned bit trick (monotonic after -0 -> +0)
__device__ __forceinline__ void atomicMaxF(float* addr, float val) {
    val += 0.f;  // canonicalize -0 to +0
    if (val >= 0.f) atomicMax((int*)addr, __float_as_int(val));
    else            atomicMin((unsigned int*)addr, __float_as_uint(val));
}

__device__ __forceinline__ v8f wmma4(v2f a, v2f b, v8f c) {
    // (neg_a, A, neg_b, B, c_mod, C, reuse_a, reuse_b)
    return __builtin_amdgcn_wmma_f32_16x16x4_f32(false, a, false, b,
                                                 (short)0, c, false, false);
}

// ------------------------------------------------------------- fp32 WMMA GEMM
// D[M,N] = A[M,K] @ B[K,N].  One wave per 16x64 output strip (4 WMMA tiles),
// K in steps of 4, software-pipelined: next step's A (1x b64) and B pairs
// (8x b32) are issued before the current step's 4 WMMAs, so load latency
// overlaps the matrix ops (partial s_wait_loadcnt instead of wait-to-zero).
// Requires M%16==0, K%8==0, N%64==0 (true: M=30000, K in {8,128}, N in
// {128,256}).
// A 16x4 layout : lane l<16 holds A[m=l][K0..K1]; lane l>=16 holds K2..K3.
// C/D 16x16     : vgpr i, lane half 0 -> row i, half 1 -> row i+8; col = lane&15.
__global__ void wmma_gemm_f32(const float* __restrict__ A,
                              const float* __restrict__ B,
                              float* __restrict__ D,
                              int M, int K, int N) {
    int wave = (blockIdx.x * blockDim.x + threadIdx.x) >> 5;
    int lane = threadIdx.x & 31;
    int ntn = N >> 6;                         // 64-col strips
    int ntm = M >> 4;
    if (wave >= ntm * ntn) return;            // whole-wave uniform exit
    int tm = wave / ntn;
    int tn = wave - tm * ntn;
    int half  = lane >> 4;
    int l15   = lane & 15;
    int half2 = half * 2;
    int bcol  = tn * 64 + l15;

    const float* aptr = A + (size_t)(tm * 16 + l15) * K + half2;
    const float* bptr = B + (size_t)half2 * N + bcol;

    // prologue: load K-step 0
    v2f a = *(const v2f*)aptr;
    v2f b0, b1, b2, b3;
    b0.x = bptr[0];  b0.y = bptr[N];
    b1.x = bptr[16]; b1.y = bptr[N + 16];
    b2.x = bptr[32]; b2.y = bptr[N + 32];
    b3.x = bptr[48]; b3.y = bptr[N + 48];

    v8f c0 = {}, c1 = {}, c2 = {}, c3 = {};
    for (int k0 = 4; k0 < K; k0 += 4) {
        aptr += 4;
        bptr += (size_t)4 * N;
        // issue next step's loads first ...
        v2f an = *(const v2f*)aptr;
        v2f n0, n1, n2, n3;
        n0.x = bptr[0];  n0.y = bptr[N];
        n1.x = bptr[16]; n1.y = bptr[N + 16];
        n2.x = bptr[32]; n2.y = bptr[N + 32];
        n3.x = bptr[48]; n3.y = bptr[N + 48];
        // ... then consume the previous step's registers
        c0 = wmma4(a, b0, c0);
        c1 = wmma4(a, b1, c1);
        c2 = wmma4(a, b2, c2);
        c3 = wmma4(a, b3, c3);
        a = an; b0 = n0; b1 = n1; b2 = n2; b3 = n3;
    }
    // epilogue: last K-step
    c0 = wmma4(a, b0, c0);
    c1 = wmma4(a, b1, c1);
    c2 = wmma4(a, b2, c2);
    c3 = wmma4(a, b3, c3);

    float* drow = D + (size_t)(tm * 16 + half * 8) * N + bcol;
#pragma unroll
    for (int i = 0; i < 8; ++i) {
        float* p = drow + (size_t)i * N;
        p[0]  = c0[i];
        p[16] = c1[i];
        p[32] = c2[i];
        p[48] = c3[i];
    }
}

// ------------------------------------------------------------- GAT edge phase

// el/er: per-(node,head) attention projections
__global__ void attn_scores(const float* __restrict__ feat,
                            const float* __restrict__ al,
                            const float* __restrict__ ar,
                            float* __restrict__ el, float* __restrict__ er,
                            int N, int H, int D) {
    int idx = blockIdx.x * blockDim.x + threadIdx.x;
    if (idx >= N * H) return;
    int n = idx / H, h = idx - n * H;
    const float* f  = feat + (size_t)n * H * D + h * D;
    const float* pl = al + h * D;
    const float* pr = ar + h * D;
    float sl = 0.f, sr = 0.f;
    for (int d = 0; d < D; ++d) { float v = f[d]; sl += v * pl[d]; sr += v * pr[d]; }
    el[idx] = sl;
    er[idx] = sr;
}

__global__ void edge_max(const float* __restrict__ el, const float* __restrict__ er,
                         const int* __restrict__ src, const int* __restrict__ dst,
                         float* __restrict__ emax, int E, int H) {
    int e = blockIdx.x * blockDim.x + threadIdx.x;
    if (e >= E) return;
    int s = src[e], d = dst[e];
    for (int h = 0; h < H; ++h)
        atomicMaxF(&emax[d * H + h], lrelu(el[s * H + h] + er[d * H + h]));
}

__global__ void edge_exp(const float* __restrict__ el, const float* __restrict__ er,
                         const int* __restrict__ src, const int* __restrict__ dst,
                         const float* __restrict__ emax, float* __restrict__ ex,
                         float* __restrict__ denom, int E, int H) {
    int e = blockIdx.x * blockDim.x + threadIdx.x;
    if (e >= E) return;
    int s = src[e], d = dst[e];
    for (int h = 0; h < H; ++h) {
        float v = lrelu(el[s * H + h] + er[d * H + h]);
        float x = expf(v - emax[d * H + h]);
        ex[e * H + h] = x;
        atomicAdd(&denom[d * H + h], x);
    }
}

// one thread per (edge, channel); consecutive lanes = consecutive channels of
// one edge -> coalesced gather of feat[src] row, atomics hit one out[dst] row.
__global__ void edge_aggregate(const float* __restrict__ feat,
                               const int* __restrict__ src,
                               const int* __restrict__ dst,
                               const float* __restrict__ ex,
                               const float* __restrict__ denom,
                               float* __restrict__ out,
                               int E, int H, int Fshift) {
    int idx = blockIdx.x * blockDim.x + threadIdx.x;
    if (idx >= (E << Fshift)) return;
    int F = 1 << Fshift;
    int e = idx >> Fshift;
    int f = idx & (F - 1);
    int h = f >> (Fshift - 2);               // D = F/4 -> h = f / D
    int s = src[e], d = dst[e];
    float alpha = ex[e * H + h] / denom[d * H + h];
    atomicAdd(&out[((size_t)d << Fshift) + f],
              feat[((size_t)s << Fshift) + f] * alpha);
}

__global__ void bias_act(float* __restrict__ out, const float* __restrict__ b,
                         int NF, int Fmask, int do_relu) {
    int idx = blockIdx.x * blockDim.x + threadIdx.x;
    if (idx >= NF) return;
    float v = out[idx] + b[idx & Fmask];
    out[idx] = do_relu ? fmaxf(v, 0.f) : v;
}

// ----------------------------------------------------------------- pooling

__global__ void pool_count(const int* __restrict__ gid, float* __restrict__ cnt, int N) {
    int n = blockIdx.x * blockDim.x + threadIdx.x;
    if (n < N) atomicAdd(&cnt[gid[n]], 1.f);
}

__global__ void pool_sum(const float* __restrict__ h, const int* __restrict__ gid,
                         float* __restrict__ hg, int N, int F) {
    int idx = blockIdx.x * blockDim.x + threadIdx.x;
    if (idx >= N * F) return;
    int n = idx / F, f = idx - n * F;
    atomicAdd(&hg[gid[n] * F + f], h[idx]);
}

__global__ void pool_div(const float* __restrict__ hg, const float* __restrict__ cnt,
                         float* __restrict__ out, int GF, int F) {
    int idx = blockIdx.x * blockDim.x + threadIdx.x;
    if (idx >= GF) return;
    int g = idx / F;
    out[idx] = hg[idx] / fmaxf(cnt[g], 1.f);
}

// ----------------------------------------------------------------- host side

static inline int cdiv(long long a, int b) { return (int)((a + b - 1) / b); }

static void run_gat_layer(const float* h, int K, int F,
                          const float* W, const float* al, const float* ar,
                          const float* b, const int* src, const int* dst,
                          float* featbuf, float* out,
                          float* el, float* er, float* emax, float* denom, float* ex,
                          int do_relu, hipStream_t stream) {
    const int H = 4;
    const int D = F / 4;
    const int Fshift = (F == 128) ? 7 : 8;

    long long waves = (long long)(NN / 16) * (F / 64);
    wmma_gemm_f32<<<cdiv(waves * 32, 256), 256, 0, stream>>>(h, W, featbuf, NN, K, F);

    attn_scores<<<cdiv((long long)NN * H, 256), 256, 0, stream>>>(featbuf, al, ar, el, er, NN, H, D);
    fill_kernel<<<cdiv((long long)NN * H, 256), 256, 0, stream>>>(emax, -INFINITY, NN * H);
    fill_kernel<<<cdiv((long long)NN * H, 256), 256, 0, stream>>>(denom, 0.f, NN * H);
    edge_max<<<cdiv(NE, 256), 256, 0, stream>>>(el, er, src, dst, emax, NE, H);
    edge_exp<<<cdiv(NE, 256), 256, 0, stream>>>(el, er, src, dst, emax, ex, denom, NE, H);

    // zero accumulator AFTER the GEMM has consumed `h` (out may alias h)
    fill_kernel<<<cdiv((long long)NN * F, 256), 256, 0, stream>>>(out, 0.f, NN * F);
    edge_aggregate<<<cdiv((long long)NE << Fshift, 256), 256, 0, stream>>>(
        featbuf, src, dst, ex, denom, out, NE, H, Fshift);
    bias_act<<<cdiv((long long)NN * F, 256), 256, 0, stream>>>(out, b, NN * F, F - 1, do_relu);
}

extern "C" void kernel_launch(void* const* d_in, const int* in_sizes, int n_in,
                              void* d_out, int out_size, void* d_ws, size_t ws_size,
                              hipStream_t stream) {
    const float* feat = (const float*)d_in[0];
    const int*   src  = (const int*)d_in[1];
    const int*   dst  = (const int*)d_in[2];
    const int*   gid  = (const int*)d_in[3];
    const float* W1 = (const float*)d_in[4];
    const float* al1 = (const float*)d_in[5];
    const float* ar1 = (const float*)d_in[6];
    const float* b1 = (const float*)d_in[7];
    const float* W2 = (const float*)d_in[8];
    const float* al2 = (const float*)d_in[9];
    const float* ar2 = (const float*)d_in[10];
    const float* b2 = (const float*)d_in[11];
    const float* W3 = (const float*)d_in[12];
    const float* al3 = (const float*)d_in[13];
    const float* ar3 = (const float*)d_in[14];
    const float* b3 = (const float*)d_in[15];

    // workspace layout (floats): ~72 MB total
    float* ws      = (float*)d_ws;
    float* featbuf = ws;                               // NN*256
    float* hbuf    = featbuf + (size_t)NN * 256;       // NN*256
    float* el      = hbuf + (size_t)NN * 256;          // NN*4
    float* er      = el + (size_t)NN * 4;              // NN*4
    float* emax    = er + (size_t)NN * 4;              // NN*4
    float* denom   = emax + (size_t)NN * 4;            // NN*4
    float* ex      = denom + (size_t)NN * 4;           // NE*4
    float* hg      = ex + (size_t)NE * 4;              // NG*256
    float* cnt     = hg + (size_t)NG * 256;            // NG
    float* h0pad   = cnt + NG;                         // NN*8

    // pad layer-1 input to K=8 so the WMMA GEMM has no K-tail predication
    pad_k6_to_k8<<<cdiv((long long)NN * 8, 256), 256, 0, stream>>>(feat, h0pad, NN);

    // layer 1: [NN,8(pad)] -> [NN,128], relu
    run_gat_layer(h0pad, 8, 128, W1, al1, ar1, b1, src, dst,
                  featbuf, hbuf, el, er, emax, denom, ex, 1, stream);
    // layer 2: [NN,128] -> [NN,128], relu (in/out alias hbuf; ordering is safe)
    run_gat_layer(hbuf, 128, 128, W2, al2, ar2, b2, src, dst,
                  featbuf, hbuf, el, er, emax, denom, ex, 1, stream);
    // layer 3: [NN,128] -> [NN,256], no relu
    run_gat_layer(hbuf, 128, 256, W3, al3, ar3, b3, src, dst,
                  featbuf, hbuf, el, er, emax, denom, ex, 0, stream);

    // mean pooling per graph -> d_out [64,256]
    fill_kernel<<<cdiv(NG * 256 + NG, 256), 256, 0, stream>>>(hg, 0.f, NG * 256 + NG);
    pool_count<<<cdiv(NN, 256), 256, 0, stream>>>(gid, cnt, NN);
    pool_sum<<<cdiv((long long)NN * 256, 256), 256, 0, stream>>>(hbuf, gid, hg, NN, 256);
    pool_div<<<cdiv(NG * 256, 256), 256, 0, stream>>>(hg, cnt, (float*)d_out, NG * 256, 256);
}